// RFCM_loss_45655502356637
// MI455X (gfx1250) — compile-verified
//
#include <hip/hip_runtime.h>
#include <math.h>

typedef __attribute__((ext_vector_type(2))) float v2f;
typedef __attribute__((ext_vector_type(8))) float v8f;

#define B_   8
#define K_   8
#define C_   3
#define H_   512
#define W_   512
#define P_   (H_ * W_)
#define REG_WT_ 0.0008f
#define TS   16
#define TW   (TS + 2)          // 18: tile + halo
#define TP   (TS + 3)          // 19: padded row pitch in LDS

#define AS1 __attribute__((address_space(1)))
#define AS3 __attribute__((address_space(3)))

// d_ws layout (floats):
//   [0 .. 31]  : accum[k*4 + c]  (c=0..2 -> num[k][c], c=3 -> den[k])
//   [32 .. 55] : v[k*3 + c]      (final centers)

__global__ void rfcm_zero_kernel(float* __restrict__ ws, float* __restrict__ out) {
    int t = threadIdx.x;
    if (t < 32) ws[t] = 0.0f;
    if (t == 0) out[0] = 0.0f;
}

// -------------------------------------------------------------------------
// Pass 1: num[K,C] and den[K] via V_WMMA_F32_16X16X4_F32, 8 pixels / WMMA.
//   A (16x4): rows 0-7  = mem_k at pixels p+0..p+3   (group 0)
//             rows 8-15 = mem_k at pixels p+4..p+7   (group 1)
//   B (4x16): cols 0-2 = img channels (group 0), col 3 = 1.0 (den),
//             cols 4-6 = img channels (group 1), col 7 = 1.0, rest 0.
//   D: block00 D[k][c] and block11 D[8+k][4+c] are the valid partials;
//      off-diagonal blocks mix the two pixel groups and are discarded.
//   All 32 lanes stream one contiguous 8B A-load per iteration.
// -------------------------------------------------------------------------
__global__ void rfcm_reduce_centers_kernel(const float* __restrict__ y_pred,
                                           const float* __restrict__ image,
                                           float* __restrict__ ws) {
    const int lane = threadIdx.x & 31;
    const int m    = lane & 15;          // A row / B column
    const int hi   = lane >> 4;          // which K-dim pair {0,1} vs {2,3}
    const int krow = m & 7;              // cluster index for A load
    const int agrp = m >> 3;             // pixel group for A row

    // B-operand role for column n = m
    const bool ldImg = (m < 3) || (m >= 4 && m < 7);
    const bool ones  = (m == 3) || (m == 7);
    const int  ch    = (m < 4) ? m : (m - 4);   // img channel when ldImg
    const int  bgrp  = (m < 4) ? 0 : 1;         // pixel group for B column

    const int wavesPerBlk = blockDim.x >> 5;
    const int wave   = blockIdx.x * wavesPerBlk + (threadIdx.x >> 5);
    const int nwaves = gridDim.x * wavesPerBlk;

    const int GPB     = P_ / 8;          // 8-pixel groups per batch image
    const int ngroups = B_ * GPB;
    const int per     = (ngroups + nwaves - 1) / nwaves;
    int g0 = wave * per;
    int g1 = g0 + per;
    if (g1 > ngroups) g1 = ngroups;

    v8f acc = {0.f, 0.f, 0.f, 0.f, 0.f, 0.f, 0.f, 0.f};

    int g = g0;
    while (g < g1) {
        const int b    = g / GPB;
        int gEnd = (b + 1) * GPB;
        if (gEnd > g1) gEnd = g1;
        const int p0   = (g - b * GPB) * 8;

        const float* ap = y_pred + ((size_t)b * K_ + krow) * P_ + p0 + 4 * agrp + 2 * hi;
        const float* ip = image  + ((size_t)b * C_ + (ldImg ? ch : 0)) * P_ + p0 + 4 * bgrp + 2 * hi;

        for (; g < gEnd; ++g, ap += 8, ip += 8) {
            float x = ap[0], y = ap[1];
            v2f a;
            a.x = x * x;                 // mem = y_pred^2
            a.y = y * y;
            v2f bv;
            if (ldImg)      { bv.x = ip[0]; bv.y = ip[1]; }
            else if (ones)  { bv.x = 1.0f;  bv.y = 1.0f;  }
            else            { bv.x = 0.0f;  bv.y = 0.0f;  }
            acc = __builtin_amdgcn_wmma_f32_16x16x4_f32(
                /*neg_a=*/false, a, /*neg_b=*/false, bv,
                /*c_mod=*/(short)0, acc, /*reuse_a=*/false, /*reuse_b=*/false);
        }
    }

    // Flush: block00 -> lanes 0..3 (M=r, N=lane); block11 -> lanes 20..23 (M=8+r, N=4+c)
    int col = -1;
    if (lane < 4)                       col = lane;
    else if (lane >= 20 && lane < 24)   col = lane - 20;
    if (col >= 0) {
        for (int r = 0; r < K_; ++r)
            atomicAdd(&ws[r * 4 + col], acc[r]);
    }
}

// Pass 2: v = num/den with forced-center override (k forced iff K-1-k < n_forced)
__global__ void rfcm_centers_kernel(const float* __restrict__ centers,
                                    float* __restrict__ ws, int n_forced) {
    int t = threadIdx.x;
    if (t < K_ * C_) {
        int k = t / C_, c = t % C_;
        float v = ws[k * 4 + c] / ws[k * 4 + 3];
        int r = K_ - 1 - k;
        if (r < n_forced) {
            int fidx = r;
            if (fidx < 0) fidx = 0;
            if (fidx > n_forced - 1) fidx = n_forced - 1;
            v = centers[fidx * C_ + c];
        }
        ws[32 + t] = v;
    }
}

// -------------------------------------------------------------------------
// Pass 3: fused J1 + J2 with double-buffered async global->LDS tile loads.
//   per pixel:  J1 = sum_k mem_k * ||img - v_k||
//               J2 = Mtot*NBtot - sum_k mem_k*nb_k   (nb = 3x3 neighbor sum)
//   Tiles hold raw y_pred; squaring happens at consume time.
// -------------------------------------------------------------------------
__device__ __forceinline__ void rfcm_issue_tile(const float* __restrict__ src,
                                                float* lbuf, int tid,
                                                int ty0, int tx0) {
    for (int idx = tid; idx < TW * TW; idx += 256) {
        const int ly = idx / TW;
        const int lx = idx - ly * TW;
        const int sy = ty0 + ly - 1;
        const int sx = tx0 + lx - 1;
        float* dst = &lbuf[ly * TP + lx];
        if (sy >= 0 && sy < H_ && sx >= 0 && sx < W_) {
            const float* gp = src + ((size_t)sy * W_ + sx);
#if __has_builtin(__builtin_amdgcn_global_load_async_to_lds_b32)
            __builtin_amdgcn_global_load_async_to_lds_b32(
                (AS1 int*)gp, (AS3 int*)dst, /*offset=*/0, /*cpol=*/0);
#else
            *dst = *gp;
#endif
        } else {
            *dst = 0.0f;                 // zero padding for the conv halo
        }
    }
}

__device__ __forceinline__ void rfcm_wait_async() {
#if __has_builtin(__builtin_amdgcn_s_wait_asynccnt)
    __builtin_amdgcn_s_wait_asynccnt(0);
#else
    asm volatile("s_wait_asynccnt 0x0" ::: "memory");
#endif
}

__global__ void rfcm_loss_kernel(const float* __restrict__ y_pred,
                                 const float* __restrict__ image,
                                 const float* __restrict__ ws,
                                 float* __restrict__ out) {
    __shared__ float tile[2][TW * TP];
    __shared__ float vsh[K_][C_];
    __shared__ float red[256];

    const int tid = threadIdx.x;
    const int tx = tid & (TS - 1);
    const int ty = tid >> 4;

    const int tilesX      = W_ / TS;                 // 32
    const int tilesPerImg = (H_ / TS) * tilesX;      // 1024
    const int b   = blockIdx.x / tilesPerImg;
    const int t   = blockIdx.x - b * tilesPerImg;
    const int tyi = t / tilesX;
    const int ty0 = tyi * TS;
    const int tx0 = (t - tyi * tilesX) * TS;
    const int gx = tx0 + tx;
    const int gy = ty0 + ty;
    const size_t pix = (size_t)gy * W_ + gx;

    if (tid < K_ * C_) vsh[tid / C_][tid % C_] = ws[32 + tid];

    const float i0 = image[((size_t)b * C_ + 0) * P_ + pix];
    const float i1 = image[((size_t)b * C_ + 1) * P_ + pix];
    const float i2 = image[((size_t)b * C_ + 2) * P_ + pix];

    const float* ybase = y_pred + (size_t)b * K_ * P_;

    // Prologue: start tile k=0 into buffer 0
    rfcm_issue_tile(ybase, tile[0], tid, ty0, tx0);

    float Mtot = 0.f, NBtot = 0.f, dotMN = 0.f, J1 = 0.f;

    for (int k = 0; k < K_; ++k) {
        rfcm_wait_async();               // my async writes for buffer (k&1) landed
        __syncthreads();                 // everyone's writes landed (also covers vsh)

        if (k + 1 < K_)
            rfcm_issue_tile(ybase + (size_t)(k + 1) * P_, tile[(k + 1) & 1],
                            tid, ty0, tx0);

        const float* tl = tile[k & 1];
        #define TAT(yy, xx) (tl[(yy) * TP + (xx)])
        float t00 = TAT(ty,     tx), t01 = TAT(ty,     tx + 1), t02 = TAT(ty,     tx + 2);
        float t10 = TAT(ty + 1, tx), t11 = TAT(ty + 1, tx + 1), t12 = TAT(ty + 1, tx + 2);
        float t20 = TAT(ty + 2, tx), t21 = TAT(ty + 2, tx + 1), t22 = TAT(ty + 2, tx + 2);
        #undef TAT

        float mk = t11 * t11;            // mem = y_pred^2
        float nb = t00 * t00 + t01 * t01 + t02 * t02
                 + t10 * t10             + t12 * t12
                 + t20 * t20 + t21 * t21 + t22 * t22;

        Mtot  += mk;
        NBtot += nb;
        dotMN += mk * nb;

        float d0 = i0 - vsh[k][0];
        float d1 = i1 - vsh[k][1];
        float d2 = i2 - vsh[k][2];
        J1 += mk * sqrtf(d0 * d0 + d1 * d1 + d2 * d2);

        __syncthreads();                 // all reads of tile[k&1] done before k+2 overwrites
    }

    float local = J1 + REG_WT_ * (Mtot * NBtot - dotMN);
    red[tid] = local;
    __syncthreads();
    for (int s = 128; s > 0; s >>= 1) {
        if (tid < s) red[tid] += red[tid + s];
        __syncthreads();
    }
    if (tid == 0)
        atomicAdd(out, red[0] * (1.0f / ((float)B_ * (float)P_)));
}

extern "C" void kernel_launch(void* const* d_in, const int* in_sizes, int n_in,
                              void* d_out, int out_size, void* d_ws, size_t ws_size,
                              hipStream_t stream) {
    const float* y_pred  = (const float*)d_in[0];   // [8,8,512,512] f32
    const float* image   = (const float*)d_in[1];   // [8,3,512,512] f32
    const float* centers = (const float*)d_in[2];   // [2,3] f32
    float* out = (float*)d_out;                     // scalar f32
    float* ws  = (float*)d_ws;

    int n_forced = in_sizes[2] / C_;                // = 2

    rfcm_zero_kernel<<<1, 64, 0, stream>>>(ws, out);

    // 512 blocks * 8 waves = 4096 waves; 262144 groups -> 64 contiguous groups/wave
    rfcm_reduce_centers_kernel<<<512, 256, 0, stream>>>(y_pred, image, ws);

    rfcm_centers_kernel<<<1, 32, 0, stream>>>(centers, ws, n_forced);

    rfcm_loss_kernel<<<B_ * (H_ / TS) * (W_ / TS), 256, 0, stream>>>(y_pred, image, ws, out);
}